// RankingMSELoss_5592047420127
// MI455X (gfx1250) — compile-verified
//
#include <hip/hip_runtime.h>

// RankingMSELoss: out = 0.7*mean((p-t)^2) + 0.3 * sum_{i<j} pair_loss / (n(n-1)/2)
// pair_loss symmetric under (i,j) swap and zero on diagonal, so we sum ALL
// ordered pairs and divide by n(n-1).
//
// Compute-bound (O(n^2) VALU f32, ~67M pairs); memory traffic negligible
// (inputs 64KB, tiled working set ~1.5MB, L2 = 192MB). WMMA is inapplicable:
// the per-pair sign/relu/abs nonlinearity cannot be phrased as D=A*B+C.
// CDNA5 paths used: wave32 blocks, GLOBAL_LOAD_ASYNC_TO_LDS_B128 staging
// (ASYNCcnt + s_wait_asynccnt), global_prefetch_b8, ds_load_b128 inner loop.

#define MARGIN 0.5f
#define MSE_W  0.7f
#define RANK_W 0.3f

constexpr int BLOCK        = 256;                  // 8 wave32s
constexpr int I_PER_THREAD = 2;
constexpr int I_TILE       = BLOCK * I_PER_THREAD; // 512 rows per block
constexpr int J_TILE       = 1024;                 // LDS-staged columns per block

// b128 async payload type, matching the builtin's parameter type exactly:
// 'int __attribute__((vector_size(16)))'
typedef int v4i_t __attribute__((vector_size(16)));
typedef __attribute__((address_space(1))) v4i_t glb_v4i_t;
typedef __attribute__((address_space(3))) v4i_t lds_v4i_t;

__device__ __forceinline__ glb_v4i_t* to_glb(const void* p) {
  return (glb_v4i_t*)(unsigned long long)p;
}
__device__ __forceinline__ lds_v4i_t* to_lds(const void* p) {
  // generic pointer to __shared__: low 32 bits == LDS byte offset (ISA 10.2)
  return (lds_v4i_t*)(unsigned int)(unsigned long long)p;
}

__device__ __forceinline__ float pair_loss(float pi, float ti, float pj, float tj) {
  float dp    = pi - pj;
  float dt    = ti - tj;
  float sdp   = (dt > 0.0f) ? dp : -dp;            // sign(dt)*dp for dt != 0
  float hinge = fmaxf(MARGIN - sdp, 0.0f);
  return (dt != 0.0f) ? hinge : fabsf(dp);         // tie -> |dp|
}

__global__ __launch_bounds__(BLOCK)
void rankmse_pair_kernel(const float* __restrict__ pred,
                         const float* __restrict__ target,
                         float* __restrict__ ws, int n) {
  __shared__ __align__(16) float sp[J_TILE];
  __shared__ __align__(16) float st[J_TILE];
  __shared__ float red[BLOCK];

  const int tid  = threadIdx.x;
  const int i0   = blockIdx.x * I_TILE;
  const int j0   = blockIdx.y * J_TILE;
  const int rem  = n - j0;
  const int jlim = rem < J_TILE ? rem : J_TILE;

  __builtin_prefetch(pred + i0, 0, 0);             // -> global_prefetch_b8
  __builtin_prefetch(target + i0, 0, 0);

  // ---- Stage j-tile (pred, target) into LDS ----
#if defined(__gfx1250__) && __has_builtin(__builtin_amdgcn_global_load_async_to_lds_b128)
  {
    const int k = tid * 4;                         // 256 threads * 16B = 4KB per array
    if (k + 4 <= jlim) {
      __builtin_amdgcn_global_load_async_to_lds_b128(to_glb(pred   + j0 + k), to_lds(&sp[k]), 0, 0);
      __builtin_amdgcn_global_load_async_to_lds_b128(to_glb(target + j0 + k), to_lds(&st[k]), 0, 0);
    } else {
      #pragma unroll
      for (int u = 0; u < 4; ++u) {
        int idx = k + u;
        if (idx < J_TILE) {
          bool ok = (j0 + idx) < n;
          sp[idx] = ok ? pred[j0 + idx]   : 0.0f;
          st[idx] = ok ? target[j0 + idx] : 0.0f;
        }
      }
    }
#if __has_builtin(__builtin_amdgcn_s_wait_asynccnt)
    __builtin_amdgcn_s_wait_asynccnt(0);
#else
    asm volatile("s_wait_asynccnt 0" ::: "memory");
#endif
  }
#else
  for (int k = tid; k < J_TILE; k += BLOCK) {
    bool ok = (j0 + k) < n;
    sp[k] = ok ? pred[j0 + k]   : 0.0f;
    st[k] = ok ? target[j0 + k] : 0.0f;
  }
#endif
  __syncthreads();

  // ---- i rows held in registers ----
  float pi[I_PER_THREAD], ti[I_PER_THREAD];
  bool  iok[I_PER_THREAD];
  #pragma unroll
  for (int r = 0; r < I_PER_THREAD; ++r) {
    int i  = i0 + r * BLOCK + tid;
    iok[r] = (i < n);
    pi[r]  = iok[r] ? pred[i]   : 0.0f;
    ti[r]  = iok[r] ? target[i] : 0.0f;
  }

  // ---- pair loop: 2 ds_load_b128 per 8 pair evaluations ----
  float acc[I_PER_THREAD];
  #pragma unroll
  for (int r = 0; r < I_PER_THREAD; ++r) acc[r] = 0.0f;

  if (jlim == J_TILE) {
    // Hot path (always taken for n % J_TILE == 0): compile-time bound, deep unroll.
    #pragma unroll 4
    for (int j = 0; j < J_TILE; j += 4) {
      float4 pj = *reinterpret_cast<const float4*>(sp + j);
      float4 tj = *reinterpret_cast<const float4*>(st + j);
      #pragma unroll
      for (int r = 0; r < I_PER_THREAD; ++r) {
        acc[r] += pair_loss(pi[r], ti[r], pj.x, tj.x);
        acc[r] += pair_loss(pi[r], ti[r], pj.y, tj.y);
        acc[r] += pair_loss(pi[r], ti[r], pj.z, tj.z);
        acc[r] += pair_loss(pi[r], ti[r], pj.w, tj.w);
      }
    }
  } else {
    const int jv = jlim & ~3;
    for (int j = 0; j < jv; j += 4) {
      float4 pj = *reinterpret_cast<const float4*>(sp + j);
      float4 tj = *reinterpret_cast<const float4*>(st + j);
      #pragma unroll
      for (int r = 0; r < I_PER_THREAD; ++r) {
        acc[r] += pair_loss(pi[r], ti[r], pj.x, tj.x);
        acc[r] += pair_loss(pi[r], ti[r], pj.y, tj.y);
        acc[r] += pair_loss(pi[r], ti[r], pj.z, tj.z);
        acc[r] += pair_loss(pi[r], ti[r], pj.w, tj.w);
      }
    }
    for (int j = jv; j < jlim; ++j) {
      #pragma unroll
      for (int r = 0; r < I_PER_THREAD; ++r)
        acc[r] += pair_loss(pi[r], ti[r], sp[j], st[j]);
    }
  }

  float accsum = 0.0f;
  #pragma unroll
  for (int r = 0; r < I_PER_THREAD; ++r) accsum += iok[r] ? acc[r] : 0.0f;

  // ---- block reduction (fixed order -> deterministic) ----
  red[tid] = accsum;
  __syncthreads();
  for (int s = BLOCK / 2; s > 0; s >>= 1) {
    if (tid < s) red[tid] += red[tid + s];
    __syncthreads();
  }
  if (tid == 0) ws[blockIdx.y * gridDim.x + blockIdx.x] = red[0];

  // ---- MSE partials: only the y==0 row of blocks touches each i once ----
  if (blockIdx.y == 0) {
    float m = 0.0f;
    #pragma unroll
    for (int r = 0; r < I_PER_THREAD; ++r) {
      float d = pi[r] - ti[r];
      m += iok[r] ? d * d : 0.0f;
    }
    __syncthreads();
    red[tid] = m;
    __syncthreads();
    for (int s = BLOCK / 2; s > 0; s >>= 1) {
      if (tid < s) red[tid] += red[tid + s];
      __syncthreads();
    }
    if (tid == 0) ws[gridDim.x * gridDim.y + blockIdx.x] = red[0];
  }
}

__global__ void rankmse_finalize_kernel(const float* __restrict__ ws,
                                        float* __restrict__ out,
                                        int n, int npair, int nmse) {
  if (threadIdx.x != 0 || blockIdx.x != 0) return;
  double pair = 0.0;
  for (int i = 0; i < npair; ++i) pair += (double)ws[i];
  double mse = 0.0;
  for (int i = 0; i < nmse; ++i) mse += (double)ws[npair + i];
  mse /= (double)n;
  if (n < 2) { out[0] = (float)mse; return; }
  // pair sums ALL ordered pairs; ranking = 0.5*pair / (n(n-1)/2) = pair/(n(n-1))
  double ranking = pair / ((double)n * (double)(n - 1));
  out[0] = (float)((double)MSE_W * mse + (double)RANK_W * ranking);
}

extern "C" void kernel_launch(void* const* d_in, const int* in_sizes, int n_in,
                              void* d_out, int out_size, void* d_ws, size_t ws_size,
                              hipStream_t stream) {
  const float* pred   = (const float*)d_in[0];
  const float* target = (const float*)d_in[1];
  float* out = (float*)d_out;
  float* ws  = (float*)d_ws;
  const int n = in_sizes[0];

  const int gx = (n + I_TILE - 1) / I_TILE;   // 16 for n=8192
  const int gy = (n + J_TILE - 1) / J_TILE;   // 8  for n=8192
  dim3 grid(gx, gy);

  rankmse_pair_kernel<<<grid, BLOCK, 0, stream>>>(pred, target, ws, n);
  rankmse_finalize_kernel<<<1, 32, 0, stream>>>(ws, out, n, gx * gy, gx);
}